// SSIMLoss_23278722744860
// MI455X (gfx1250) — compile-verified
//
#include <hip/hip_runtime.h>

typedef __attribute__((ext_vector_type(2))) float v2f;
typedef __attribute__((ext_vector_type(8))) float v8f;

#define IMG_H 512
#define IMG_W 512
#define TILE 16
#define HALO 5
#define PATCH 26          // TILE + 2*HALO
#define PSTRIDE 27        // padded patch row stride (bank-conflict free)
#define HSTRIDE 18        // padded H row stride
#define NWAVES 4
#define PATCH_FLOATS (PATCH * PSTRIDE)                 // 702
#define HMAP_FLOATS (PATCH * HSTRIDE)                  // 468
#define WAVE_LDS_FLOATS (2 * PATCH_FLOATS + 5 * HMAP_FLOATS)  // 3744
#define C1_SSIM 1.0e-4f
#define C2_SSIM 9.0e-4f
#define N_TILES (48 * 32 * 32)     // 49152
#define N_BLOCKS (N_TILES / NWAVES)

// Normalized 11-tap Gaussian, sigma = 1.5
__constant__ float GW[11] = {
    0.00102838f, 0.00759876f, 0.03600077f, 0.10936010f, 0.21300589f,
    0.26601210f, 0.21300589f, 0.10936010f, 0.03600077f, 0.00759876f,
    0.00102838f};

static __device__ __forceinline__ v8f wmma4(v2f a, v2f b, v8f c) {
  // D(16x16,f32) = A(16x4,f32) * B(4x16,f32) + C
  return __builtin_amdgcn_wmma_f32_16x16x4_f32(false, a, false, b, (short)0, c,
                                               false, false);
}

__global__ __launch_bounds__(NWAVES * 32) void ssim_tile_kernel(
    const float* __restrict__ pred, const float* __restrict__ targ,
    float* __restrict__ partial) {
  __shared__ float lds[NWAVES * WAVE_LDS_FLOATS];
  __shared__ float block_red[NWAVES];

  const int wave = threadIdx.x >> 5;
  const int lane = threadIdx.x & 31;
  const int half = lane >> 4;      // K half: lanes 0-15 -> K {0,1}, 16-31 -> {2,3}
  const int nm = lane & 15;        // M (A rows) or N (B cols / D cols) for this lane

  float* __restrict__ P = lds + wave * WAVE_LDS_FLOATS;
  float* __restrict__ T = P + PATCH_FLOATS;
  float* __restrict__ Hm = T + PATCH_FLOATS;  // 5 maps of 26x16 (stride 18)

  const int tile = blockIdx.x * NWAVES + wave;
  const int img = tile >> 10;           // 1024 tiles per 512x512 plane
  const int rem = tile & 1023;
  const int ty0 = (rem >> 5) << 4;
  const int tx0 = (rem & 31) << 4;
  const float* pp = pred + (size_t)img * (IMG_H * IMG_W);
  const float* tt = targ + (size_t)img * (IMG_H * IMG_W);

  // Stage 26x26 halo patch of pred and target into LDS (zero pad = SAME conv).
  for (int i = lane; i < PATCH * PATCH; i += 32) {
    int r = i / PATCH;
    int c = i - r * PATCH;
    int gy = ty0 - HALO + r;
    int gx = tx0 - HALO + c;
    float pv = 0.0f, tv = 0.0f;
    if (gy >= 0 && gy < IMG_H && gx >= 0 && gx < IMG_W) {
      int g = gy * IMG_W + gx;
      pv = pp[g];
      tv = tt[g];
    }
    P[r * PSTRIDE + c] = pv;
    T[r * PSTRIDE + c] = tv;
  }
  __syncthreads();

  // Per-lane banded Gaussian fragment: element value g[c - nm] at the K-slot
  // positions this lane owns (c = 4*k4 + 2*half + {0,1}).  Serves as the B
  // operand in the horizontal pass and as the A operand in the vertical pass
  // (f32 A and B layouts mirror each other across lanes/halves).
  v2f wfrag[7];
#pragma unroll
  for (int k4 = 0; k4 < 7; ++k4) {
    int c0 = k4 * 4 + 2 * half;
    int d0 = c0 - nm;
    wfrag[k4].x = ((unsigned)d0 <= 10u) ? GW[d0] : 0.0f;
    wfrag[k4].y = ((unsigned)(d0 + 1) <= 10u) ? GW[d0 + 1] : 0.0f;
  }

  // ---- Horizontal blur: H[r][n] = sum_c patch[r][c] * g[c-n], r in 0..25 ----
#pragma unroll
  for (int rg = 0; rg < 2; ++rg) {
    const int rowbase = rg * 16;
    v8f a0 = {}, a1 = {}, a2 = {}, a3 = {}, a4 = {};
#pragma unroll
    for (int k4 = 0; k4 < 7; ++k4) {
      const int c0 = k4 * 4 + 2 * half;
      const int row = rowbase + nm;
      const bool rok = row < PATCH;
      float p0 = (rok && c0 < PATCH) ? P[row * PSTRIDE + c0] : 0.0f;
      float p1 = (rok && (c0 + 1) < PATCH) ? P[row * PSTRIDE + c0 + 1] : 0.0f;
      float t0 = (rok && c0 < PATCH) ? T[row * PSTRIDE + c0] : 0.0f;
      float t1 = (rok && (c0 + 1) < PATCH) ? T[row * PSTRIDE + c0 + 1] : 0.0f;
      v2f a;
      a.x = p0;      a.y = p1;      a0 = wmma4(a, wfrag[k4], a0);
      a.x = t0;      a.y = t1;      a1 = wmma4(a, wfrag[k4], a1);
      a.x = p0 * p0; a.y = p1 * p1; a2 = wmma4(a, wfrag[k4], a2);
      a.x = t0 * t0; a.y = t1 * t1; a3 = wmma4(a, wfrag[k4], a3);
      a.x = p0 * t0; a.y = p1 * t1; a4 = wmma4(a, wfrag[k4], a4);
    }
    // D layout: lane holds rows v + 8*half, col nm.  Keep rows < 26.
#pragma unroll
    for (int v = 0; v < 8; ++v) {
      int row = rowbase + v + 8 * half;
      if (row < PATCH) {
        Hm[0 * HMAP_FLOATS + row * HSTRIDE + nm] = a0[v];
        Hm[1 * HMAP_FLOATS + row * HSTRIDE + nm] = a1[v];
        Hm[2 * HMAP_FLOATS + row * HSTRIDE + nm] = a2[v];
        Hm[3 * HMAP_FLOATS + row * HSTRIDE + nm] = a3[v];
        Hm[4 * HMAP_FLOATS + row * HSTRIDE + nm] = a4[v];
      }
    }
  }
  __syncthreads();

  // ---- Vertical blur: out[y][x] = sum_c g[c-y] * H[c][x] ----
  v8f vmu1 = {}, vmu2 = {}, vp2 = {}, vt2 = {}, vpt = {};
#pragma unroll
  for (int k4 = 0; k4 < 7; ++k4) {
    const int c0 = k4 * 4 + 2 * half;
    const bool ok0 = c0 < PATCH;
    const bool ok1 = (c0 + 1) < PATCH;
    v2f b;
    b.x = ok0 ? Hm[0 * HMAP_FLOATS + c0 * HSTRIDE + nm] : 0.0f;
    b.y = ok1 ? Hm[0 * HMAP_FLOATS + (c0 + 1) * HSTRIDE + nm] : 0.0f;
    vmu1 = wmma4(wfrag[k4], b, vmu1);
    b.x = ok0 ? Hm[1 * HMAP_FLOATS + c0 * HSTRIDE + nm] : 0.0f;
    b.y = ok1 ? Hm[1 * HMAP_FLOATS + (c0 + 1) * HSTRIDE + nm] : 0.0f;
    vmu2 = wmma4(wfrag[k4], b, vmu2);
    b.x = ok0 ? Hm[2 * HMAP_FLOATS + c0 * HSTRIDE + nm] : 0.0f;
    b.y = ok1 ? Hm[2 * HMAP_FLOATS + (c0 + 1) * HSTRIDE + nm] : 0.0f;
    vp2 = wmma4(wfrag[k4], b, vp2);
    b.x = ok0 ? Hm[3 * HMAP_FLOATS + c0 * HSTRIDE + nm] : 0.0f;
    b.y = ok1 ? Hm[3 * HMAP_FLOATS + (c0 + 1) * HSTRIDE + nm] : 0.0f;
    vt2 = wmma4(wfrag[k4], b, vt2);
    b.x = ok0 ? Hm[4 * HMAP_FLOATS + c0 * HSTRIDE + nm] : 0.0f;
    b.y = ok1 ? Hm[4 * HMAP_FLOATS + (c0 + 1) * HSTRIDE + nm] : 0.0f;
    vpt = wmma4(wfrag[k4], b, vpt);
  }

  // ---- SSIM map (all five accumulators share the same C/D layout) ----
  float lsum = 0.0f;
#pragma unroll
  for (int v = 0; v < 8; ++v) {
    float m1 = vmu1[v], m2 = vmu2[v];
    float m1s = m1 * m1, m2s = m2 * m2, m12 = m1 * m2;
    float s1 = vp2[v] - m1s;
    float s2 = vt2[v] - m2s;
    float s12 = vpt[v] - m12;
    float num = (2.0f * m12 + C1_SSIM) * (2.0f * s12 + C2_SSIM);
    float den = (m1s + m2s + C1_SSIM) * (s1 + s2 + C2_SSIM);
    lsum += num / den;
  }
#pragma unroll
  for (int off = 16; off > 0; off >>= 1) lsum += __shfl_xor(lsum, off, 32);
  if (lane == 0) block_red[wave] = lsum;
  __syncthreads();
  if (threadIdx.x == 0) {
    float s = 0.0f;
#pragma unroll
    for (int w = 0; w < NWAVES; ++w) s += block_red[w];
    partial[blockIdx.x] = s;
  }
}

__global__ __launch_bounds__(256) void ssim_finalize_kernel(
    const float* __restrict__ partial, int n, float* __restrict__ out,
    float inv_count) {
  __shared__ float red[256];
  float s = 0.0f;
  for (int i = threadIdx.x; i < n; i += 256) s += partial[i];
  red[threadIdx.x] = s;
  __syncthreads();
  for (int stride = 128; stride > 0; stride >>= 1) {
    if (threadIdx.x < stride) red[threadIdx.x] += red[threadIdx.x + stride];
    __syncthreads();
  }
  if (threadIdx.x == 0) out[0] = 1.0f - red[0] * inv_count;
}

extern "C" void kernel_launch(void* const* d_in, const int* in_sizes, int n_in,
                              void* d_out, int out_size, void* d_ws,
                              size_t ws_size, hipStream_t stream) {
  (void)in_sizes; (void)n_in; (void)out_size; (void)ws_size;
  const float* pred = (const float*)d_in[0];
  const float* targ = (const float*)d_in[1];
  float* partial = (float*)d_ws;  // N_BLOCKS floats (~48 KB)

  ssim_tile_kernel<<<N_BLOCKS, NWAVES * 32, 0, stream>>>(pred, targ, partial);
  ssim_finalize_kernel<<<1, 256, 0, stream>>>(
      partial, N_BLOCKS, (float*)d_out,
      1.0f / (16.0f * 3.0f * 512.0f * 512.0f));
}